// DDKFLayer_17540646437424
// MI455X (gfx1250) — compile-verified
//
#include <hip/hip_runtime.h>
#include <math.h>

typedef float v2f __attribute__((ext_vector_type(2)));
typedef float v8f __attribute__((ext_vector_type(8)));

#define N_I   3997      // interpolated length
#define NW    995       // number of windows
#define NSIG  1000
#define NB    4
#define WIN   20
#define WT    63        // window tiles of 16 (63*16 = 1008 >= 995)
#define KT    250       // k tiles of 16   (250*16 = 4000 >= 3997)
#define TWO_PI 6.283185307179586f

// workspace layout (float offsets)
#define OFF_KS    0        // NB*N_I              = 15988
#define OFF_TW    16000    // float2[N_I]         = 7994 floats
#define OFF_ERE   32000    // WIN*N_I             = 79940
#define OFF_EIM   112000   // WIN*N_I             = 79940
#define OFF_F     192000   // float2[NB*N_I]      = 31976 floats
#define OFF_RMAX  224000   // NB*NW               = 3980
#define OFF_BMAX  228000   // NB uints

// ---------------- Pass 1: cubic interpolation + kernel mix -> ks ----------------
__global__ void k_interp(const float* __restrict__ sig,
                         const float* __restrict__ gamma_raw,
                         float* __restrict__ ws) {
  __shared__ float interp[N_I];
  __shared__ float red[256];
  const int b = blockIdx.x;
  const int tid = threadIdx.x;
  const float* s = sig + b * NSIG;
  const float A = -0.75f;
  float lmin = 3.4e38f;
  for (int p = tid; p < N_I; p += 256) {
    int i0 = p >> 2;
    float t = 0.25f * (float)(p & 3);     // s = pos - i0, exact
    float d0 = t + 1.0f;
    float w0 = ((A * d0 - 5.0f * A) * d0 + 8.0f * A) * d0 - 4.0f * A;   // w_far(s+1)
    float w1 = ((A + 2.0f) * t - (A + 3.0f)) * t * t + 1.0f;            // w_near(s)
    float t1 = 1.0f - t;
    float w2 = ((A + 2.0f) * t1 - (A + 3.0f)) * t1 * t1 + 1.0f;         // w_near(1-s)
    float d3 = 2.0f - t;
    float w3 = ((A * d3 - 5.0f * A) * d3 + 8.0f * A) * d3 - 4.0f * A;   // w_far(2-s)
    int im1 = (i0 - 1 < 0) ? 0 : i0 - 1;
    int ip1 = (i0 + 1 > NSIG - 1) ? NSIG - 1 : i0 + 1;
    int ip2 = (i0 + 2 > NSIG - 1) ? NSIG - 1 : i0 + 2;
    float v = w0 * s[im1] + w1 * s[i0] + w2 * s[ip1] + w3 * s[ip2];
    interp[p] = v;
    lmin = fminf(lmin, v);
  }
  red[tid] = lmin;
  __syncthreads();
  for (int o = 128; o > 0; o >>= 1) {
    if (tid < o) red[tid] = fminf(red[tid], red[tid + o]);
    __syncthreads();
  }
  float mshift = fminf(red[0], 0.0f);
  float r0 = gamma_raw[0], r1 = gamma_raw[1];
  float mx = fmaxf(r0, r1);
  float e0 = expf(r0 - mx), e1 = expf(r1 - mx);
  float inv = 1.0f / (e0 + e1);
  float g0 = e0 * inv, g1 = e1 * inv;
  float* ks = ws + OFF_KS + b * N_I;
  for (int p = tid; p < N_I; p += 256) {
    float x = interp[p] - mshift;
    float poly = (x + 1.3f) * (x + 1.3f);
    float dg = x - 0.7f;
    float gauss = expf(-0.5f * dg * dg);
    ks[p] = g0 * poly + g1 * gauss;
  }
}

// ---------------- Pass 2a: twiddle + E tables, zero batch maxes ----------------
__global__ void k_tables(float* __restrict__ ws) {
  int gid = blockIdx.x * blockDim.x + threadIdx.x;
  const float step = -(TWO_PI / (float)N_I);
  if (gid < N_I) {
    float sn, cs;
    __sincosf(step * (float)gid, &sn, &cs);   // e^{-2*pi*i*m/n}
    ((float2*)(ws + OFF_TW))[gid] = make_float2(cs, sn);
  }
  if (gid < WIN * N_I) {
    int j = gid / N_I;
    int k = gid - j * N_I;
    unsigned m = ((unsigned)j * (unsigned)k) % (unsigned)N_I;
    float sn, cs;
    __sincosf(step * (float)m, &sn, &cs);
    ws[OFF_ERE + gid] = cs;
    ws[OFF_EIM + gid] = sn;
  }
  if (gid < NB) ((unsigned*)(ws + OFF_BMAX))[gid] = 0u;
}

// ---------------- Pass 2b: full direct DFT per batch -> F_full ----------------
__global__ void k_dft(float* __restrict__ ws) {
  __shared__ float2 stw[N_I];   // ~32 KB
  const int tid = threadIdx.x;
  const float2* twg = (const float2*)(ws + OFF_TW);
  for (int m = tid; m < N_I; m += 256) stw[m] = twg[m];
  __syncthreads();
  int gid = blockIdx.x * 256 + tid;
  if (gid >= NB * N_I) return;
  int b = gid / N_I;
  int k = gid - b * N_I;
  const float* ks = ws + OFF_KS + b * N_I;
  float re = 0.0f, im = 0.0f;
  unsigned m = 0;
  for (int p = 0; p < N_I; ++p) {
    float2 t = stw[m];
    float v = ks[p];
    re = fmaf(v, t.x, re);
    im = fmaf(v, t.y, im);
    m += (unsigned)k;
    if (m >= (unsigned)N_I) m -= (unsigned)N_I;
  }
  ((float2*)(ws + OFF_F))[gid] = make_float2(re, im);
}

// ---------------- Pass 3a: WMMA windowed-DFT magnitude M + row maxima ----------------
__global__ void k_win_mag(float* __restrict__ ws, float* __restrict__ out) {
  const int wt = blockIdx.x;           // window tile
  const int b  = blockIdx.y;           // batch
  const int tid = threadIdx.x;         // 128 threads = 4 waves
  const int wave = tid >> 5;
  const int lane = tid & 31;
  const int half = lane >> 4;
  const int lrow = lane & 15;
  const int w0 = wt * 16;

  const float* ks  = ws + OFF_KS + b * N_I;
  const float* Ere = ws + OFF_ERE;
  const float* Eim = ws + OFF_EIM;

  // A tile: A[row = lane&15][K]; VGPR0 holds K = 4kk + 2*half, VGPR1 K+1
  v2f a[5];
  const int wa = w0 + lrow;
  const bool wvalid = (wa < NW);
  #pragma unroll
  for (int kk = 0; kk < 5; ++kk) {
    int j = 4 * kk + 2 * half;
    a[kk].x = wvalid ? ks[4 * wa + j] : 0.0f;
    a[kk].y = wvalid ? ks[4 * wa + j + 1] : 0.0f;
  }

  float rmax[8];
  #pragma unroll
  for (int i = 0; i < 8; ++i) rmax[i] = 0.0f;

  for (int t = wave; t < KT; t += 4) {
    const int k0 = t * 16;
    const int col = k0 + lrow;
    const int colc = (col < N_I) ? col : (N_I - 1);
    __builtin_prefetch(Ere + ((t + 4 < KT) ? (t + 4) * 16 : 0) + lrow, 0, 1);
    v8f cre = {};
    v8f cim = {};
    #pragma unroll
    for (int kk = 0; kk < 5; ++kk) {
      int j = 4 * kk + 2 * half;
      v2f br, bi;
      br.x = Ere[j * N_I + colc];
      br.y = Ere[(j + 1) * N_I + colc];
      bi.x = Eim[j * N_I + colc];
      bi.y = Eim[(j + 1) * N_I + colc];
      cre = __builtin_amdgcn_wmma_f32_16x16x4_f32(false, a[kk], false, br, (short)0, cre, false, false);
      cim = __builtin_amdgcn_wmma_f32_16x16x4_f32(false, a[kk], false, bi, (short)0, cim, false, false);
    }
    #pragma unroll
    for (int i = 0; i < 8; ++i) {
      int w = w0 + i + 8 * half;                 // C/D: row = vgpr + 8*(lane>=16)
      float M = sqrtf(cre[i] * cre[i] + cim[i] * cim[i]);
      bool ok = (w < NW) && (col < N_I);
      if (ok) out[((size_t)b * NW + w) * (size_t)N_I + col] = M;
      rmax[i] = fmaxf(rmax[i], ok ? M : 0.0f);
    }
  }

  __shared__ float red[4][16][16];
  #pragma unroll
  for (int i = 0; i < 8; ++i) red[wave][i + 8 * half][lrow] = rmax[i];
  __syncthreads();
  if (tid < 16) {
    float m = 0.0f;
    for (int v = 0; v < 4; ++v)
      for (int c = 0; c < 16; ++c) m = fmaxf(m, red[v][tid][c]);
    int w = w0 + tid;
    if (w < NW) ws[OFF_RMAX + b * NW + w] = m;
  }
}

// ---------------- Pass 3b: recompute G, rotate, M1, masked result + batch max ----------------
__global__ void k_win_result(float* __restrict__ ws, float* __restrict__ out,
                             const float* __restrict__ alpha_p) {
  const int wt = blockIdx.x;
  const int b  = blockIdx.y;
  const int tid = threadIdx.x;
  const int wave = tid >> 5;
  const int lane = tid & 31;
  const int half = lane >> 4;
  const int lrow = lane & 15;
  const int w0 = wt * 16;

  const float alpha = *alpha_p;
  const float* ks  = ws + OFF_KS + b * N_I;
  const float* Ere = ws + OFF_ERE;
  const float* Eim = ws + OFF_EIM;
  const float2* tw = (const float2*)(ws + OFF_TW);
  const float2* Ff = ((const float2*)(ws + OFF_F)) + b * N_I;
  const float* rowmax = ws + OFF_RMAX + b * NW;

  v2f a[5];
  const int wa = w0 + lrow;
  const bool wvalid = (wa < NW);
  #pragma unroll
  for (int kk = 0; kk < 5; ++kk) {
    int j = 4 * kk + 2 * half;
    a[kk].x = wvalid ? ks[4 * wa + j] : 0.0f;
    a[kk].y = wvalid ? ks[4 * wa + j + 1] : 0.0f;
  }

  float lmax = 0.0f;
  for (int t = wave; t < KT; t += 4) {
    const int k0 = t * 16;
    const int col = k0 + lrow;
    const int colc = (col < N_I) ? col : (N_I - 1);
    v8f cre = {};
    v8f cim = {};
    #pragma unroll
    for (int kk = 0; kk < 5; ++kk) {
      int j = 4 * kk + 2 * half;
      v2f br, bi;
      br.x = Ere[j * N_I + colc];
      br.y = Ere[(j + 1) * N_I + colc];
      bi.x = Eim[j * N_I + colc];
      bi.y = Eim[(j + 1) * N_I + colc];
      cre = __builtin_amdgcn_wmma_f32_16x16x4_f32(false, a[kk], false, br, (short)0, cre, false, false);
      cim = __builtin_amdgcn_wmma_f32_16x16x4_f32(false, a[kk], false, bi, (short)0, cim, false, false);
    }
    #pragma unroll
    for (int i = 0; i < 8; ++i) {
      int w = w0 + i + 8 * half;
      if (w < NW && col < N_I) {
        size_t idx = ((size_t)b * NW + w) * (size_t)N_I + col;
        float Mv = out[idx];                                     // M from pass 3a
        unsigned m = (4u * (unsigned)w * (unsigned)col) % (unsigned)N_I;
        float2 r = tw[m];                                        // e^{-2pi i 4wk/n}
        float fre = r.x * cre[i] - r.y * cim[i];                 // F_win = rot * G
        float fim = r.x * cim[i] + r.y * cre[i];
        float2 F = Ff[col];
        float dre = F.x - fre, dim2 = F.y - fim;                 // F1 = F_full - F_win
        float M1 = sqrtf(dre * dre + dim2 * dim2);
        float strong = (Mv > rowmax[w] * alpha) ? 1.0f : 0.0f;
        float comb = M1 * Mv * strong;
        float res = fminf(Mv, comb);
        out[idx] = res;
        lmax = fmaxf(lmax, res);
      }
    }
  }

  __shared__ float red[128];
  red[tid] = lmax;
  __syncthreads();
  for (int o = 64; o > 0; o >>= 1) {
    if (tid < o) red[tid] = fmaxf(red[tid], red[tid + o]);
    __syncthreads();
  }
  if (tid == 0)
    atomicMax(((unsigned int*)(ws + OFF_BMAX)) + b, __float_as_uint(red[0]));
}

// ---------------- Pass 4: global threshold ----------------
__global__ void k_thresh(float* __restrict__ out, const float* __restrict__ ws,
                         const float* __restrict__ beta_p) {
  size_t idx = (size_t)blockIdx.x * 256 + threadIdx.x;
  const size_t total = (size_t)NB * NW * N_I;
  if (idx >= total) return;
  int b = (int)(idx / ((size_t)NW * N_I));
  float bm = __uint_as_float(((const unsigned*)(ws + OFF_BMAX))[b]);
  float thr = (*beta_p) * bm;
  float v = out[idx];
  out[idx] = (v > thr) ? v : 0.0f;
}

extern "C" void kernel_launch(void* const* d_in, const int* in_sizes, int n_in,
                              void* d_out, int out_size, void* d_ws, size_t ws_size,
                              hipStream_t stream) {
  const float* sig       = (const float*)d_in[0];  // (4,1000)
  const float* alpha     = (const float*)d_in[1];  // scalar
  const float* beta      = (const float*)d_in[2];  // scalar
  const float* gamma_raw = (const float*)d_in[3];  // (2,)
  float* out = (float*)d_out;
  float* ws  = (float*)d_ws;

  hipLaunchKernelGGL(k_interp, dim3(NB), dim3(256), 0, stream, sig, gamma_raw, ws);

  int ntab = WIN * N_I;
  hipLaunchKernelGGL(k_tables, dim3((ntab + 255) / 256), dim3(256), 0, stream, ws);

  hipLaunchKernelGGL(k_dft, dim3((NB * N_I + 255) / 256), dim3(256), 0, stream, ws);

  hipLaunchKernelGGL(k_win_mag, dim3(WT, NB), dim3(128), 0, stream, ws, out);

  hipLaunchKernelGGL(k_win_result, dim3(WT, NB), dim3(128), 0, stream, ws, out, alpha);

  size_t total = (size_t)NB * NW * N_I;
  hipLaunchKernelGGL(k_thresh, dim3((unsigned)((total + 255) / 256)), dim3(256), 0, stream,
                     out, ws, beta);
}